// Router_54932631716286
// MI455X (gfx1250) — compile-verified
//
#include <hip/hip_runtime.h>

typedef __attribute__((ext_vector_type(2))) float v2f;
typedef __attribute__((ext_vector_type(8))) float v8f;

#define DIM 768
#define NEXP 8
#define WSTRIDE 772                     // 768 + 4 pad -> conflict-free LDS B reads
#define TOK_PER_WAVE 16
#define WAVES_PER_BLOCK 8
#define TOK_PER_BLOCK (TOK_PER_WAVE * WAVES_PER_BLOCK)  // 128

__global__ __launch_bounds__(256, 1) void router_wmma_kernel(
    const float* __restrict__ x,        // [ntok, DIM]
    const float* __restrict__ W,        // [NEXP, DIM]
    const float* __restrict__ bias,     // [NEXP]
    float* __restrict__ out_gates,      // [ntok, 2]
    float* __restrict__ out_idx,        // [ntok, 2] (indices stored as float)
    int ntok)
{
    __shared__ float sW[16 * WSTRIDE];              // W zero-padded to 16 experts
    __shared__ float sLog[TOK_PER_BLOCK * NEXP];    // logits tile

    const int tid = threadIdx.x;

    // ---- Stage W (8x768) into LDS, zero-pad rows 8..15 and the stride pad ----
    for (int i = tid; i < 16 * WSTRIDE; i += 256) {
        int r = i / WSTRIDE;
        int col = i - r * WSTRIDE;
        sW[i] = (r < NEXP && col < DIM) ? W[r * DIM + col] : 0.0f;
    }
    __syncthreads();

    const int wave = tid >> 5;
    const int lane = tid & 31;
    const int half = lane >> 4;     // 0: lanes 0-15, 1: lanes 16-31
    const int mn   = lane & 15;     // A fragment: row M; B fragment: col N
    const int koff = half * 2;      // K sub-offset inside the K=4 slab

    const int tile_base = blockIdx.x * TOK_PER_BLOCK + wave * TOK_PER_WAVE;
    // Clamp so EXEC stays all-ones (WMMA requirement); OOB results never stored.
    int row_tok = tile_base + mn;
    if (row_tok >= ntok) row_tok = ntok - 1;
    const float* xrow = x + (size_t)row_tok * DIM;
    const float* wrow = &sW[mn * WSTRIDE];

    // ---- 16x16 logits tile via V_WMMA_F32_16X16X4_F32 over K=768 ----
    v8f c = {0.f, 0.f, 0.f, 0.f, 0.f, 0.f, 0.f, 0.f};
    #pragma unroll 8
    for (int k = 0; k < DIM; k += 4) {
        v2f a = *(const v2f*)(xrow + k + koff);       // A[M=mn][K=k+koff, +1]
        v2f bfrag = *(const v2f*)(wrow + k + koff);   // B[K][N=mn] = W[N][k+K]
        // 8 args: (neg_a, A, neg_b, B, c_mod, C, reuse_a, reuse_b)
        c = __builtin_amdgcn_wmma_f32_16x16x4_f32(
                false, a, false, bfrag, (short)0, c, false, false);
    }

    // ---- Scatter C tile: c[v] = logit[M = v + 8*half][N = mn] ----
    if (mn < NEXP) {
        #pragma unroll
        for (int v = 0; v < 8; ++v) {
            int m = v + half * 8;
            sLog[(wave * TOK_PER_WAVE + m) * NEXP + mn] = c[v];
        }
    }
    __syncthreads();

    // ---- Per-token bias + softmax + top-2 (threads 0..127, one token each) ----
    if (tid < TOK_PER_BLOCK) {
        int token = blockIdx.x * TOK_PER_BLOCK + tid;
        if (token < ntok) {
            float l[NEXP];
            #pragma unroll
            for (int e = 0; e < NEXP; ++e)
                l[e] = sLog[tid * NEXP + e] + bias[e];

            // top-1 (strict > : lowest index wins ties, matches lax.top_k)
            float m1 = l[0]; int i1 = 0;
            #pragma unroll
            for (int e = 1; e < NEXP; ++e)
                if (l[e] > m1) { m1 = l[e]; i1 = e; }
            // top-2
            float m2 = -3.402823466e+38f; int i2 = 0;
            #pragma unroll
            for (int e = 0; e < NEXP; ++e)
                if (e != i1 && l[e] > m2) { m2 = l[e]; i2 = e; }

            // softmax denominator (max-subtracted); top gates only
            float sum = 0.0f;
            #pragma unroll
            for (int e = 0; e < NEXP; ++e)
                sum += __expf(l[e] - m1);
            float inv = 1.0f / sum;
            float g1 = inv;                       // exp(0)/sum
            float g2 = __expf(m2 - m1) * inv;

            out_gates[token * 2 + 0] = g1;
            out_gates[token * 2 + 1] = g2;
            out_idx[token * 2 + 0] = (float)i1;
            out_idx[token * 2 + 1] = (float)i2;
        }
    }
}

extern "C" void kernel_launch(void* const* d_in, const int* in_sizes, int n_in,
                              void* d_out, int out_size, void* d_ws, size_t ws_size,
                              hipStream_t stream) {
    const float* x = (const float*)d_in[0];
    const float* W = (const float*)d_in[1];
    const float* b = (const float*)d_in[2];
    float* out = (float*)d_out;

    int ntok = in_sizes[0] / DIM;                 // 128*197 = 25216
    int nblocks = (ntok + TOK_PER_BLOCK - 1) / TOK_PER_BLOCK;  // 197

    float* out_gates = out;
    float* out_idx   = out + (size_t)ntok * 2;

    router_wmma_kernel<<<nblocks, 256, 0, stream>>>(x, W, b, out_gates, out_idx, ntok);
}